// ARMA2d_72705206387355
// MI455X (gfx1250) — compile-verified
//
#include <hip/hip_runtime.h>
#include <math.h>

typedef float v2f __attribute__((ext_vector_type(2)));
typedef float v8f __attribute__((ext_vector_type(8)));

#define PI2_64 0.0981747704246810387f  /* 2*pi/64 */
#define CIP 258                        /* padded ci stride in conv LDS tile */

// ---------------------------------------------------------------------------
// Per-channel inverse circulant filter rows:
//   3-tap circular kernel along an axis: a(-1)=h0, a(0)=1, a(+1)=h2
//   F(m) = 1 + (h0+h2)cos(th_m) + i (h0-h2) sin(th_m)
//   g[n] = (1/64) * sum_m Re( e^{+i th_m n} / F(m) )
// ---------------------------------------------------------------------------
__global__ void arma_filters(const float* __restrict__ alpha,
                             float* __restrict__ gh, float* __restrict__ gw) {
  int c = blockIdx.x;       // 256 channels
  int n = threadIdx.x;      // 64 taps of the inverse kernel
  const float cth = 0.70710678118654752f;   // cos(-pi/4)
  const float sth = -0.70710678118654752f;  // sin(-pi/4)
  float t00 = tanhf(alpha[c * 4 + 0]);
  float t01 = tanhf(alpha[c * 4 + 1]);
  float t10 = tanhf(alpha[c * 4 + 2]);
  float t11 = tanhf(alpha[c * 4 + 3]);
  float h0 = t00 * cth - t01 * sth, h2 = t00 * sth + t01 * cth;
  float w0 = t10 * cth - t11 * sth, w2 = t10 * sth + t11 * cth;
  float hs = h0 + h2, hd = h0 - h2;
  float ws_ = w0 + w2, wd = w0 - w2;
  float sh = 0.f, sw = 0.f;
  for (int m = 0; m < 64; ++m) {
    float th = PI2_64 * (float)m;
    float cm = cosf(th), sm = sinf(th);
    int mn = (m * n) & 63;                 // angle reduction: (m*n) mod 64
    float an = PI2_64 * (float)mn;
    float cn = cosf(an), sn = sinf(an);
    { float re = 1.f + hs * cm, im = hd * sm;
      sh += (cn * re + sn * im) / (re * re + im * im); }
    { float re = 1.f + ws_ * cm, im = wd * sm;
      sw += (cn * re + sn * im) / (re * re + im * im); }
  }
  gh[c * 64 + n] = sh * (1.0f / 64.0f);
  gw[c * 64 + n] = sw * (1.0f / 64.0f);
}

// ---------------------------------------------------------------------------
// Repack conv weights into exact WMMA A-fragment order:
//   wt2[((tap*64 + chunk)*16 + cot)*32 + lane] = v2f{ w[co][ci], w[co][ci+1] }
//   with co = cot*16 + (lane&15), ci = chunk*4 + 2*(lane>>4)
// -> one coalesced global_load_b64 per fragment in the conv kernel.
// ---------------------------------------------------------------------------
__global__ void repack_w(const float* __restrict__ w, float* __restrict__ wt2) {
  int idx = blockIdx.x * 256 + threadIdx.x;   // 9*64*16*32 = 294912 fragments
  if (idx >= 9 * 64 * 16 * 32) return;
  int lane = idx & 31;
  int rest = idx >> 5;
  int cot = rest & 15;  rest >>= 4;
  int chunk = rest & 63;
  int tap = rest >> 6;
  int m = lane & 15;
  int kh = 2 * (lane >> 4);
  int co = cot * 16 + m;
  int ci = chunk * 4 + kh;
  v2f v;
  v.x = w[((size_t)co * 256 + ci) * 9 + tap];
  v.y = w[((size_t)co * 256 + ci + 1) * 9 + tap];
  ((v2f*)wt2)[idx] = v;
}

// ---------------------------------------------------------------------------
// Implicit-GEMM 3x3 conv, V_WMMA_F32_16X16X4_F32.
// One block (512 thr = 16 waves) computes a 32-pixel row strip for ALL 256 co:
//  - stage 3x34x256 input halo in LDS once (boundary zeros via cndmask,
//    no exec-mask branching), layout [row][col][ci] (ci unit-stride, padded)
//  - wave w = co-tile w; each A fragment feeds TWO pixel tiles:
//    per 2 WMMAs: 1 global b64 (A) + 2 ds b64 (B0,B1); two independent
//    accumulator chains interleave in the matrix pipe.
// ---------------------------------------------------------------------------
__global__ __launch_bounds__(512)
void conv3x3_wmma(const float* __restrict__ x, const float* __restrict__ wt2,
                  float* __restrict__ y) {
  __shared__ float Xs[3 * 34 * CIP];          // ~105 KB (of 320 KB/WGP)
  int tid = threadIdx.x;
  int pix = blockIdx.x;                        // 4096 = b(32)*h(64)*strip(2)
  int strip = pix & 1;
  int h = (pix >> 1) & 63;
  int b = pix >> 7;
  int w0 = strip * 32;

  // ---- stage input halo: rows h-1..h+1, cols w0-1..w0+32, all 256 ci ----
  for (int e = tid; e < 3 * 34 * 256; e += 512) {
    int ci = e / 102;           // 102 = 3*34 (row,col) pairs per channel
    int rc = e - ci * 102;
    int row = rc / 34;
    int col = rc - row * 34;
    int hh = h + row - 1;
    int wc = w0 + col - 1;
    bool ok = ((unsigned)hh < 64u) && ((unsigned)wc < 64u);
    int hcl = min(max(hh, 0), 63);
    int wcl = min(max(wc, 0), 63);
    float v = x[(((size_t)b * 256 + ci) * 64 + hcl) * 64 + wcl];
    Xs[(row * 34 + col) * CIP + ci] = ok ? v : 0.0f;
  }
  __syncthreads();

  int cot = tid >> 5;           // wave id == co tile
  int lane = tid & 31;
  int m = lane & 15;
  int khalf = (lane >> 4) * 2;
  int co0 = cot * 16;

  const v2f* wfrag = (const v2f*)wt2;
  v8f acc0 = {};
  v8f acc1 = {};
#pragma unroll
  for (int tap = 0; tap < 9; ++tap) {
    int row = tap / 3, dx = tap - row * 3;
    const float* xb0 = &Xs[(row * 34 + m + dx) * CIP + khalf];
    const float* xb1 = xb0 + 16 * CIP;         // second pixel tile (cols +16)
    const v2f* wb = &wfrag[(size_t)(tap * 64 * 16 + cot) * 32 + lane];
    for (int chunk = 0; chunk < 64; ++chunk) {
      v2f a = wb[(size_t)chunk * 16 * 32];
      v2f b0 = *(const v2f*)&xb0[chunk * 4];
      v2f b1 = *(const v2f*)&xb1[chunk * 4];
      acc0 = __builtin_amdgcn_wmma_f32_16x16x4_f32(false, a, false, b0,
                                                   (short)0, acc0, false, false);
      acc1 = __builtin_amdgcn_wmma_f32_16x16x4_f32(false, a, false, b1,
                                                   (short)0, acc1, false, false);
    }
  }
  // C/D layout: VGPR r -> row co0 + r + 8*(lane>>4), col = m
  int rbase = co0 + ((lane >> 4) ? 8 : 0);
  size_t obase = (((size_t)b * 256 + rbase) * 64 + h) * 64 + (w0 + m);
#pragma unroll
  for (int r = 0; r < 8; ++r) y[obase + (size_t)r * 4096] = acc0[r];
#pragma unroll
  for (int r = 0; r < 8; ++r) y[obase + 16 + (size_t)r * 4096] = acc1[r];
}

// ---------------------------------------------------------------------------
// Per-(batch,channel) deconvolution (the FFT division) as two circulant
// 64x64 GEMMs:  Out = Mh * Y * Mw^T,  M[i,p] = g[(i-p) mod 64].
// ghr/gwr are unwrapped reversed tables: ghr[u] = gh[(64-u)&63], u in [0,129],
// so both circulant taps of a fragment are adjacent LDS reads (no wrap math).
// Tt holds the intermediate transposed [j][k] so stage-2 B is contiguous b64.
// ---------------------------------------------------------------------------
__global__ __launch_bounds__(256)
void deconv_wmma(const float* __restrict__ gh, const float* __restrict__ gw,
                 float* __restrict__ y) {
  __shared__ float Ys[64 * 64];
  __shared__ float Tt[64 * 64];
  __shared__ float ghr[130];
  __shared__ float gwr[130];
  int bc = blockIdx.x;          // b*256 + c
  int c = bc & 255;
  int tid = threadIdx.x;
  if (tid < 130) {
    ghr[tid] = gh[c * 64 + ((64 - tid) & 63)];
    gwr[tid] = gw[c * 64 + ((64 - tid) & 63)];
  }
  float* img = y + (size_t)bc * 4096;
  for (int i = tid; i < 4096; i += 256) Ys[i] = img[i];
  __syncthreads();

  int wave = tid >> 5, lane = tid & 31;
  int m = lane & 15;
  int khalf = (lane >> 4) * 2;

  // Stage 1: T = Y * Mw^T  (A = Y rows, B[k][j] = gw[(j-k)&63] = gwr[k-j+64])
  for (int t = wave * 2; t < wave * 2 + 2; ++t) {
    int ib = (t >> 2) * 16, jb = (t & 3) * 16;
    int i = ib + m, j = jb + m;
    v8f acc = {};
    for (int k0 = 0; k0 < 64; k0 += 4) {
      int k = k0 + khalf;
      v2f a = *(const v2f*)&Ys[i * 64 + k];     // 8B aligned (k even)
      v2f bb;
      bb.x = gwr[k - j + 64];
      bb.y = gwr[k - j + 65];
      acc = __builtin_amdgcn_wmma_f32_16x16x4_f32(false, a, false, bb,
                                                  (short)0, acc, false, false);
    }
    int rbase = ib + ((lane >> 4) ? 8 : 0);
    float* dst = &Tt[(jb + m) * 64 + rbase];    // transposed store [j][i]
#pragma unroll
    for (int r = 0; r < 8; ++r) dst[r] = acc[r];
  }
  __syncthreads();

  // Stage 2: Out = Mh * T  (A[i][p] = gh[(i-p)&63] = ghr[p-i+64], B = T)
  for (int t = wave * 2; t < wave * 2 + 2; ++t) {
    int ib = (t >> 2) * 16, jb = (t & 3) * 16;
    int i = ib + m, j = jb + m;
    v8f acc = {};
    for (int k0 = 0; k0 < 64; k0 += 4) {
      int k = k0 + khalf;
      v2f a;
      a.x = ghr[k - i + 64];
      a.y = ghr[k - i + 65];
      v2f bb = *(const v2f*)&Tt[j * 64 + k];    // 8B aligned (k even)
      acc = __builtin_amdgcn_wmma_f32_16x16x4_f32(false, a, false, bb,
                                                  (short)0, acc, false, false);
    }
    int rbase = ib + ((lane >> 4) ? 8 : 0);
#pragma unroll
    for (int r = 0; r < 8; ++r) img[(rbase + r) * 64 + jb + m] = acc[r];
  }
}

// ---------------------------------------------------------------------------
extern "C" void kernel_launch(void* const* d_in, const int* in_sizes, int n_in,
                              void* d_out, int out_size, void* d_ws, size_t ws_size,
                              hipStream_t stream) {
  const float* x = (const float*)d_in[0];      // (32,256,64,64)
  const float* w = (const float*)d_in[1];      // (256,256,3,3)
  const float* alpha = (const float*)d_in[2];  // (256,1,2,2)
  float* out = (float*)d_out;                  // (32,256,64,64)
  float* wsf = (float*)d_ws;
  float* gh = wsf;                   // 256*64
  float* gw = wsf + 256 * 64;        // 256*64
  float* wt2 = wsf + 2 * 256 * 64;   // 9*64*16*32 v2f = 2.36 MB

  arma_filters<<<256, 64, 0, stream>>>(alpha, gh, gw);
  repack_w<<<1152, 256, 0, stream>>>(w, wt2);
  // one block per 32-pixel strip; 16 waves cover all 256 output channels,
  // each A fragment feeds two 16-pixel WMMA tiles
  conv3x3_wmma<<<4096, 512, 0, stream>>>(x, wt2, out);
  // one block per (batch, channel) image, in-place on d_out
  deconv_wmma<<<8192, 256, 0, stream>>>(gh, gw, out);
}